// ProductBasisBlock_44676249813159
// MI455X (gfx1250) — compile-verified
//
#include <hip/hip_runtime.h>

typedef __attribute__((ext_vector_type(2))) float v2f;
typedef __attribute__((ext_vector_type(8))) float v8f;

#define Bsz   4096
#define Csz   256
#define LM    16

// workspace float offsets
#define OFF_W3   0          // [c][x][y][i]  : 256*16*16*16 = 1048576
#define OFF_C2   1048576    // [c][x][y]     : 256*256      = 65536
#define OFF_C1   1114112    // [c][i]        : 256*16       = 4096
#define OFF_O1   1118208    // [c][b]        : 256*4096     = 1048576 (column-major for coalesced GEMM loads)

// ---------------------------------------------------------------------------
// Kernel 1: fold U-tensors with per-channel weights (B-independent, tiny).
//   W3[c,x,y,i] = sum_k U3[x,y,i,k] * w3[k,c]      (K3 = 23)
//   C2[c,x,i]   = sum_k U2[x,i,k]   * w2[k,c]      (K2 = 8)
//   C1[c,i]     = sum_k U1[i,k]     * w1[k,c]      (K1 = 4)
// ---------------------------------------------------------------------------
__global__ __launch_bounds__(256) void pb_precompute(
    const float* __restrict__ U3, const float* __restrict__ U2,
    const float* __restrict__ U1, const float* __restrict__ w3,
    const float* __restrict__ w2, const float* __restrict__ w1,
    float* __restrict__ ws)
{
    int idx = blockIdx.x * 256 + threadIdx.x;
    if (idx < 1048576) {                      // W3
        int c   = idx >> 12;                  // [c][xyi]
        int xyi = idx & 4095;                 // x*256 + y*16 + i  == U3 row
        float s = 0.f;
        #pragma unroll
        for (int k = 0; k < 23; ++k) s += U3[xyi * 23 + k] * w3[k * Csz + c];
        ws[OFF_W3 + idx] = s;
    } else if (idx < 1048576 + 65536) {       // C2
        int e  = idx - 1048576;
        int c  = e >> 8;
        int xi = e & 255;                     // x*16 + i
        float s = 0.f;
        #pragma unroll
        for (int k = 0; k < 8; ++k) s += U2[xi * 8 + k] * w2[k * Csz + c];
        ws[OFF_C2 + c * 256 + xi] = s;
    } else if (idx < 1048576 + 65536 + 4096) { // C1
        int e = idx - 1048576 - 65536;
        int c = e >> 4;
        int i = e & 15;
        float s = 0.f;
        #pragma unroll
        for (int k = 0; k < 4; ++k) s += U1[i * 4 + k] * w1[k * Csz + c];
        ws[OFF_C1 + c * 16 + i] = s;
    }
}

// ---------------------------------------------------------------------------
// Kernel 2: main contraction. Block = 8 waves, one channel c per block,
// each wave owns a 16-row b-tile. Per x: D[y,b] = sum_i W3c[x,y,i]*nf[b,i]
// via 4 chained V_WMMA_F32_16X16X4_F32 with C preloaded with c2[c,x,y].
// Then out2[b,x] = sum_y D[y,b]*nf[b,y] (7 adds + shfl_xor 16), and
// out1[b] accumulates (out2 + c1[c,x]) * nf[b,x] over x.
// ---------------------------------------------------------------------------
__global__ __launch_bounds__(256) void pb_main(
    const float* __restrict__ nf,   // (B, C, 16)
    const float* __restrict__ ws,
    float* __restrict__ out1)       // [c][b]
{
    __shared__ float sW3[4096];     // W3[c] : [x][y][i]
    __shared__ float sC2[256];      // [x][y]
    __shared__ float sC1[16];
    __shared__ float sN[8 * 256];   // per-wave 16x16 nf tile [b_local][i]

    const int t  = threadIdx.x;
    const int c  = blockIdx.y;
    const int w  = t >> 5;
    const int l  = t & 31;
    const int l16 = l & 15;
    const int hi  = l >> 4;

    // --- stage W3[c] (16 KB) : 4 float4 per thread, lane-consecutive ---
    {
        const float4* g = (const float4*)(ws + OFF_W3 + c * 4096);
        float4* s = (float4*)sW3;
        #pragma unroll
        for (int j = 0; j < 4; ++j) s[j * 256 + t] = g[j * 256 + t];
    }
    // --- stage c2 row block + c1 ---
    sC2[t] = ws[OFF_C2 + c * 256 + t];
    if (t < 16) sC1[t] = ws[OFF_C1 + c * 16 + t];
    // --- stage this wave's nf tile: 2 float4 per thread ---
    {
        const int row  = l >> 1;
        const int half = l & 1;
        const int b    = (blockIdx.x * 8 + w) * 16 + row;
        const float4* g = (const float4*)(nf + (size_t)b * (Csz * LM) + c * LM + half * 8);
        float4* s = (float4*)(sN + w * 256 + row * 16 + half * 8);
        s[0] = g[0];
        s[1] = g[1];
    }
    __syncthreads();

    const float* Nw = sN + w * 256;

    // B operand (nf[b,i], b = lane, i = K) is x-invariant: preload 4 chunks
    v2f Bk[4];
    #pragma unroll
    for (int kk = 0; kk < 4; ++kk) {
        int i0 = 4 * kk + 2 * hi;
        Bk[kk].x = Nw[l16 * 16 + i0];
        Bk[kk].y = Nw[l16 * 16 + i0 + 1];
    }
    // nf in D layout: nfd[r] = nf[b=l16, y=r+8*hi]
    float nfd[8];
    #pragma unroll
    for (int r = 0; r < 8; ++r) nfd[r] = Nw[l16 * 16 + r + 8 * hi];

    float acc = 0.f;

    #pragma unroll 4
    for (int x = 0; x < 16; ++x) {
        // C initialized with broadcast c2[c,x,y] row (y = r + 8*hi)
        v8f C;
        #pragma unroll
        for (int r = 0; r < 8; ++r) C[r] = sC2[x * 16 + r + 8 * hi];

        const float* Ax = sW3 + x * 256;    // A[m=y][k=i] = W3c[x,y,i]
        #pragma unroll
        for (int kk = 0; kk < 4; ++kk) {
            int i0 = 4 * kk + 2 * hi;
            v2f A;
            A.x = Ax[l16 * 16 + i0];
            A.y = Ax[l16 * 16 + i0 + 1];
            C = __builtin_amdgcn_wmma_f32_16x16x4_f32(
                    false, A, false, Bk[kk], (short)0, C, false, false);
        }
        // out2[b,x]: sum over y (8 in registers + other half via shfl_xor 16)
        float e = C[0] * nfd[0];
        #pragma unroll
        for (int r = 1; r < 8; ++r) e += C[r] * nfd[r];
        e += __shfl_xor(e, 16, 32);
        // step 3 accumulate
        acc += (e + sC1[x]) * Nw[l16 * 16 + x];
    }

    if (hi == 0) {
        int b = (blockIdx.x * 8 + w) * 16 + l16;
        out1[c * Bsz + b] = acc;            // column-major [c][b] -> coalesced
    }
}

// ---------------------------------------------------------------------------
// Kernel 3: res[b,d] = (1/16) * sum_c out1T[c][b] * lw[c][d]
// M=4096, N=256, K=256 f32 GEMM, one 16x16 tile per wave, K in chunks of 4.
// ---------------------------------------------------------------------------
__global__ __launch_bounds__(256) void pb_final_gemm(
    const float* __restrict__ o1,   // [c][b] (K-major for A)
    const float* __restrict__ lw,   // (C, C)
    float* __restrict__ out)        // (B, C)
{
    const int t   = threadIdx.x;
    const int w   = t >> 5;
    const int l   = t & 31;
    const int l16 = l & 15;
    const int hi  = l >> 4;

    const int tile = blockIdx.x * 8 + w;     // 4096 tiles: 256 (M) x 16 (N)
    const int b0 = (tile >> 4) * 16;
    const int d0 = (tile & 15) * 16;

    v8f acc;
    #pragma unroll
    for (int r = 0; r < 8; ++r) acc[r] = 0.f;

    #pragma unroll 4
    for (int k0 = 0; k0 < 256; k0 += 4) {
        const int ka = k0 + 2 * hi;
        v2f A, B;
        A.x = o1[(size_t)ka * Bsz + b0 + l16];
        A.y = o1[(size_t)(ka + 1) * Bsz + b0 + l16];
        B.x = lw[ka * Csz + d0 + l16];
        B.y = lw[(ka + 1) * Csz + d0 + l16];
        acc = __builtin_amdgcn_wmma_f32_16x16x4_f32(
                  false, A, false, B, (short)0, acc, false, false);
    }

    #pragma unroll
    for (int r = 0; r < 8; ++r)
        out[(size_t)(b0 + r + 8 * hi) * Csz + d0 + l16] = acc[r] * 0.0625f;
}

// ---------------------------------------------------------------------------
extern "C" void kernel_launch(void* const* d_in, const int* in_sizes, int n_in,
                              void* d_out, int out_size, void* d_ws, size_t ws_size,
                              hipStream_t stream) {
    (void)in_sizes; (void)n_in; (void)out_size; (void)ws_size;
    const float* nf  = (const float*)d_in[0];
    const float* U3  = (const float*)d_in[1];
    const float* U2  = (const float*)d_in[2];
    const float* U1  = (const float*)d_in[3];
    const float* w3  = (const float*)d_in[4];
    const float* w2  = (const float*)d_in[5];
    const float* w1  = (const float*)d_in[6];
    const float* lw  = (const float*)d_in[7];

    float* ws   = (float*)d_ws;
    float* out1 = ws + OFF_O1;

    // 1) fold U-tensors with channel weights (1118208 elems)
    pb_precompute<<<4368, 256, 0, stream>>>(U3, U2, U1, w3, w2, w1, ws);
    // 2) main contraction: grid.x = B/(8*16) = 32 tile groups, grid.y = C
    pb_main<<<dim3(32, 256), 256, 0, stream>>>(nf, ws, out1);
    // 3) channel-mixing GEMM with 1/sqrt(C) scale
    pb_final_gemm<<<512, 256, 0, stream>>>(out1, lw, (float*)d_out);
}